// DigitCaps_28174985462321
// MI455X (gfx1250) — compile-verified
//
#include <hip/hip_runtime.h>

typedef __attribute__((ext_vector_type(2))) float v2f;
typedef __attribute__((ext_vector_type(8))) float v8f;

#define B_ 512
#define R_ 1152
#define C_ 10
#define O_ 16
#define I_ 8

// One wave computes 16(batch) x 16(o) f32 tiles for one r across all 10
// capsules via chained V_WMMA_F32_16X16X4_F32 (K=8 = 2 x k4 steps).
// A-tile (x) is loaded once and reused for all capsules; output is streamed
// with non-temporal stores using a single base pointer + immediate offsets.
__global__ __launch_bounds__(256) void digitcaps_wmma_kernel(
    const float* __restrict__ x,     // [B, R, I]
    const float* __restrict__ W,     // [R, C, O, I]
    const float* __restrict__ bias,  // [O]
    float* __restrict__ out)         // [B, R, C, O]
{
    const int lane = threadIdx.x & 31;
    const int wave = threadIdx.x >> 5;
    const int wtid = blockIdx.x * 8 + wave;  // = r*32 + mt
    const int mt = wtid & 31;                // batch tile (M)
    const int r  = wtid >> 5;                // row 0..1151
    if (r >= R_) return;                     // wave-uniform

    const int half = lane >> 4;              // 0: lanes 0-15, 1: lanes 16-31
    const int l16  = lane & 15;

    // ---- A tile: x[b, r, 0..7], M = l16 in both lane halves ----
    // lanes 0-15 hold K={0,1} (then {4,5}); lanes 16-31 hold K={2,3} ({6,7})
    const int b = mt * 16 + l16;
    const float* ap = x + ((size_t)b * R_ + r) * I_ + half * 2;
    const v2f a0 = *(const v2f*)(ap);        // K = base, base+1
    const v2f a1 = *(const v2f*)(ap + 4);    // K = base+4, base+5

    // bias[o]: C/D layout has N = l16 for both halves, and o == N here
    const float bv = bias[l16];

    const float* wr = W + (size_t)r * (C_ * O_ * I_);

    // Single per-lane output base: row (mt*16 + half*8), capsule 0, col l16.
    // All 80 stores are base + {j*R*C*O + ct*O} elements; max byte offset
    // 7*737280 + 9*64 = 5,161,536 B < 2^23, so they fold into IOFFSET.
    float* outp = out + (((size_t)(mt * 16 + half * 8) * R_ + r) * (C_ * O_)) + l16;

    #pragma unroll
    for (int ct = 0; ct < C_; ++ct) {
        // ---- B tile: W[r, ct, o=l16, i], B(k, n=o) = W_tile[o*8 + k] ----
        const float* bp = wr + ct * (O_ * I_) + l16 * I_ + half * 2;
        const v2f b0 = *(const v2f*)(bp);
        const v2f b1 = *(const v2f*)(bp + 4);

        v8f acc;
        #pragma unroll
        for (int j = 0; j < 8; ++j) acc[j] = bv;   // seed with bias[o]

        // 8 args: (neg_a, A, neg_b, B, c_mod, C, reuse_a, reuse_b)
        acc = __builtin_amdgcn_wmma_f32_16x16x4_f32(
            false, a0, false, b0, (short)0, acc, false, false);
        acc = __builtin_amdgcn_wmma_f32_16x16x4_f32(
            false, a1, false, b1, (short)0, acc, false, false);

        // ---- Store: VGPR j -> batch row (mt*16 + half*8 + j), col o=l16 ----
        #pragma unroll
        for (int j = 0; j < 8; ++j) {
            __builtin_nontemporal_store(
                acc[j], outp + (size_t)j * (R_ * C_ * O_) + ct * O_);
        }
    }
}

extern "C" void kernel_launch(void* const* d_in, const int* in_sizes, int n_in,
                              void* d_out, int out_size, void* d_ws, size_t ws_size,
                              hipStream_t stream) {
    const float* x    = (const float*)d_in[0];  // [512,1152,8]
    const float* W    = (const float*)d_in[1];  // [1,1152,10,16,8]
    const float* bias = (const float*)d_in[2];  // [16,1]
    float* out = (float*)d_out;                 // [512,1152,10,16,1]

    const int total_waves = R_ * (B_ / 16);     // 1152*32 = 36864
    const int blocks = total_waves / 8;         // 4608 blocks of 8 waves
    digitcaps_wmma_kernel<<<blocks, 256, 0, stream>>>(x, W, bias, out);
}